// DyITAModule_14637248545391
// MI455X (gfx1250) — compile-verified
//
#include <hip/hip_runtime.h>
#include <cmath>

typedef __attribute__((ext_vector_type(2))) float v2f;
typedef __attribute__((ext_vector_type(8))) float v8f;

#define Bn 8
#define Cc 320
#define Hh 64
#define Ww 64
#define Ss (Hh*Ww)        // 4096
#define Nn 256
#define NH 2
#define HD (Cc/NH)        // 160
#define Pp 3
#define Ff 9
#define BSz (Bn*Ss)       // 32768
#define BNz (Bn*Nn)       // 2048

static __device__ __forceinline__ float wave_sum(float v) {
#pragma unroll
  for (int o = 16; o > 0; o >>= 1) v += __shfl_xor(v, o, 32);
  return v;
}

// ---------------------------------------------------------------------------
// Tiled WMMA GEMM (fp32, v_wmma_f32_16x16x4_f32).
// C[z] = alpha * op(A[z]) * op(B[z]).
// A logical [M,K]: TA=0 stored [M,K]; TA=1 stored [K,M].
// B logical [K,N]: TB=0 stored [K,N]; TB=1 stored [N,K]  (X*W^T case).
// Requires: K % 32 == 0; lda/ldb 4-float multiples; 16B-aligned base pointers.
// Block tile 128(M) x 64(N), K-slab 32.  8 waves; wave tile 32x32 (4 accums).
// Batch z: offsets = (z/zdiv)*s?1 + (z%zdiv)*s?2.
// rowSel: if non-null, only rows with rowSel[row]==selVal are written.
// ---------------------------------------------------------------------------
template <int TA, int TB>
__global__ __launch_bounds__(256) void wmma_gemm(
    const float* __restrict__ A, const float* __restrict__ B, float* __restrict__ C,
    int M, int N, int K, int lda, int ldb, int ldc,
    long long sA1, long long sA2, long long sB1, long long sB2,
    long long sC1, long long sC2, int zdiv,
    float alpha, const int* __restrict__ rowSel, int selVal)
{
  __shared__ __align__(16) float As[128][36];   // [m][k], row stride 36 (16B mult, conflict-free)
  __shared__ __align__(16) float Bs[64][36];    // [n][k]

  int z = blockIdx.z;
  int zb = z / zdiv, zh = z % zdiv;
  A += zb * sA1 + zh * sA2;
  B += zb * sB1 + zh * sB2;
  C += zb * sC1 + zh * sC2;

  int tid = threadIdx.x;
  int lane = tid & 31, w = tid >> 5;
  int wm = (w & 3) << 5;           // wave M offset in tile: 0,32,64,96
  int wn = (w >> 2) << 5;          // wave N offset in tile: 0,32
  int gm0 = blockIdx.y << 7, gn0 = blockIdx.x << 6;
  bool interior = (gm0 + 128 <= M) && (gn0 + 64 <= N);

  v8f acc00 = {}, acc01 = {}, acc10 = {}, acc11 = {};
  float4 ra[4], rb[2];

  // ---- global -> register staging -------------------------------------
  auto loadA = [&](int k0) {
    if (TA == 0) {
      int r = tid >> 3, cc = (tid & 7) << 2;        // 4 consecutive k
      if (interior) {
#pragma unroll
        for (int j = 0; j < 4; ++j)
          ra[j] = *(const float4*)(A + (long long)(gm0 + r + 32 * j) * lda + k0 + cc);
      } else {
#pragma unroll
        for (int j = 0; j < 4; ++j) {
          int gm = gm0 + r + 32 * j;
          int gmc = gm < M ? gm : M - 1;
          const float* p = A + (long long)gmc * lda + k0 + cc;
          float4 v; v.x = p[0]; v.y = p[1]; v.z = p[2]; v.w = p[3];
          if (gm >= M) { v.x = 0.f; v.y = 0.f; v.z = 0.f; v.w = 0.f; }
          ra[j] = v;
        }
      }
    } else {                                         // stored [K,M]: 4 consecutive m
      int ak = tid >> 5, am = (tid & 31) << 2;
      if (interior) {
#pragma unroll
        for (int j = 0; j < 4; ++j)
          ra[j] = *(const float4*)(A + (long long)(k0 + ak + 8 * j) * lda + gm0 + am);
      } else {
#pragma unroll
        for (int j = 0; j < 4; ++j) {
          const float* row = A + (long long)(k0 + ak + 8 * j) * lda;
          int mb = gm0 + am;
          float4 v;
          int c0 = mb + 0 < M ? mb + 0 : M - 1; float t0 = row[c0];
          int c1 = mb + 1 < M ? mb + 1 : M - 1; float t1 = row[c1];
          int c2 = mb + 2 < M ? mb + 2 : M - 1; float t2 = row[c2];
          int c3 = mb + 3 < M ? mb + 3 : M - 1; float t3 = row[c3];
          v.x = (mb + 0 < M) ? t0 : 0.f;
          v.y = (mb + 1 < M) ? t1 : 0.f;
          v.z = (mb + 2 < M) ? t2 : 0.f;
          v.w = (mb + 3 < M) ? t3 : 0.f;
          ra[j] = v;
        }
      }
    }
  };
  auto loadB = [&](int k0) {
    if (TB == 1) {                                   // stored [N,K]: 4 consecutive k
      int r = tid >> 3, cc = (tid & 7) << 2;
      if (interior) {
#pragma unroll
        for (int j = 0; j < 2; ++j)
          rb[j] = *(const float4*)(B + (long long)(gn0 + r + 32 * j) * ldb + k0 + cc);
      } else {
#pragma unroll
        for (int j = 0; j < 2; ++j) {
          int gn = gn0 + r + 32 * j;
          int gnc = gn < N ? gn : N - 1;
          const float* p = B + (long long)gnc * ldb + k0 + cc;
          float4 v; v.x = p[0]; v.y = p[1]; v.z = p[2]; v.w = p[3];
          if (gn >= N) { v.x = 0.f; v.y = 0.f; v.z = 0.f; v.w = 0.f; }
          rb[j] = v;
        }
      }
    } else {                                         // stored [K,N]: 4 consecutive n
      int bk = tid >> 4, bn = (tid & 15) << 2;
      if (interior) {
#pragma unroll
        for (int j = 0; j < 2; ++j)
          rb[j] = *(const float4*)(B + (long long)(k0 + bk + 16 * j) * ldb + gn0 + bn);
      } else {
#pragma unroll
        for (int j = 0; j < 2; ++j) {
          const float* row = B + (long long)(k0 + bk + 16 * j) * ldb;
          int nb = gn0 + bn;
          float4 v;
          int c0 = nb + 0 < N ? nb + 0 : N - 1; float t0 = row[c0];
          int c1 = nb + 1 < N ? nb + 1 : N - 1; float t1 = row[c1];
          int c2 = nb + 2 < N ? nb + 2 : N - 1; float t2 = row[c2];
          int c3 = nb + 3 < N ? nb + 3 : N - 1; float t3 = row[c3];
          v.x = (nb + 0 < N) ? t0 : 0.f;
          v.y = (nb + 1 < N) ? t1 : 0.f;
          v.z = (nb + 2 < N) ? t2 : 0.f;
          v.w = (nb + 3 < N) ? t3 : 0.f;
          rb[j] = v;
        }
      }
    }
  };
  // ---- register -> LDS -------------------------------------------------
  auto storeA = [&]() {
    if (TA == 0) {
      int r = tid >> 3, cc = (tid & 7) << 2;
#pragma unroll
      for (int j = 0; j < 4; ++j) *(float4*)&As[r + 32 * j][cc] = ra[j];
    } else {
      int ak = tid >> 5, am = (tid & 31) << 2;
#pragma unroll
      for (int j = 0; j < 4; ++j) {
        int k = ak + 8 * j;
        As[am + 0][k] = ra[j].x; As[am + 1][k] = ra[j].y;
        As[am + 2][k] = ra[j].z; As[am + 3][k] = ra[j].w;
      }
    }
  };
  auto storeB = [&]() {
    if (TB == 1) {
      int r = tid >> 3, cc = (tid & 7) << 2;
#pragma unroll
      for (int j = 0; j < 2; ++j) *(float4*)&Bs[r + 32 * j][cc] = rb[j];
    } else {
      int bk = tid >> 4, bn = (tid & 15) << 2;
#pragma unroll
      for (int j = 0; j < 2; ++j) {
        int k = bk + 16 * j;
        Bs[bn + 0][k] = rb[j].x; Bs[bn + 1][k] = rb[j].y;
        Bs[bn + 2][k] = rb[j].z; Bs[bn + 3][k] = rb[j].w;
      }
    }
  };

  // ---- software-pipelined main loop -----------------------------------
  loadA(0); loadB(0);
  int kp = (lane >> 4) << 1;           // A/B frag: lanes 0-15 -> K 0,1; lanes 16-31 -> K 2,3
  int ma = wm + (lane & 15);
  int nb = wn + (lane & 15);
  for (int k0 = 0; k0 < K; k0 += 32) {
    storeA(); storeB();
    __syncthreads();
    if (k0 + 32 < K) { loadA(k0 + 32); loadB(k0 + 32); }   // prefetch next slab
#pragma unroll
    for (int kk = 0; kk < 32; kk += 4) {
      int kx = kk + kp;
      v2f a0 = *(const v2f*)&As[ma][kx];
      v2f a1 = *(const v2f*)&As[ma + 16][kx];
      v2f b0 = *(const v2f*)&Bs[nb][kx];
      v2f b1 = *(const v2f*)&Bs[nb + 16][kx];
      acc00 = __builtin_amdgcn_wmma_f32_16x16x4_f32(false, a0, false, b0, (short)0, acc00, false, false);
      acc01 = __builtin_amdgcn_wmma_f32_16x16x4_f32(false, a0, false, b1, (short)0, acc01, false, false);
      acc10 = __builtin_amdgcn_wmma_f32_16x16x4_f32(false, a1, false, b0, (short)0, acc10, false, false);
      acc11 = __builtin_amdgcn_wmma_f32_16x16x4_f32(false, a1, false, b1, (short)0, acc11, false, false);
    }
    __syncthreads();
  }

  // ---- epilogue (C/D layout: vgpr r -> M=r | r+8 per lane half; N=lane&15)
  int mb0 = gm0 + wm + ((lane >> 4) << 3);
  int gn = gn0 + wn + (lane & 15);
  bool n0ok = gn < N, n1ok = gn + 16 < N;
#pragma unroll
  for (int r = 0; r < 8; ++r) {
    int row0 = mb0 + r, row1 = row0 + 16;
    if (row0 < M && (rowSel == nullptr || rowSel[row0] == selVal)) {
      if (n0ok) C[(long long)row0 * ldc + gn]      = alpha * acc00[r];
      if (n1ok) C[(long long)row0 * ldc + gn + 16] = alpha * acc01[r];
    }
    if (row1 < M && (rowSel == nullptr || rowSel[row1] == selVal)) {
      if (n0ok) C[(long long)row1 * ldc + gn]      = alpha * acc10[r];
      if (n1ok) C[(long long)row1 * ldc + gn + 16] = alpha * acc11[r];
    }
  }
}

// ---------------------------------------------------------------------------
// Depthwise 3x3 conv + per-channel batch statistics (one block per (b,c) plane)
// ---------------------------------------------------------------------------
__global__ __launch_bounds__(256) void dwc_stats_kernel(
    const float* __restrict__ img, const float* __restrict__ wgt,
    float* __restrict__ dwc, float* __restrict__ stats)
{
  __shared__ float red[256];
  int bc = blockIdx.x;
  int c = bc % Cc;
  const float* plane = img + (long long)bc * Ss;
  float* out = dwc + (long long)bc * Ss;
  float k[9];
#pragma unroll
  for (int i = 0; i < 9; ++i) k[i] = wgt[c * 9 + i];
  float s = 0.f, s2 = 0.f;
  for (int t = threadIdx.x; t < Ss; t += 256) {
    int h = t >> 6, x = t & 63;
    float acc = 0.f;
#pragma unroll
    for (int dy = -1; dy <= 1; ++dy) {
      int hh = h + dy;
      if ((unsigned)hh >= (unsigned)Hh) continue;
#pragma unroll
      for (int dx = -1; dx <= 1; ++dx) {
        int xx = x + dx;
        if ((unsigned)xx >= (unsigned)Ww) continue;
        acc += plane[hh * Ww + xx] * k[(dy + 1) * 3 + (dx + 1)];
      }
    }
    out[t] = acc;
    s += acc; s2 += acc * acc;
  }
  red[threadIdx.x] = s; __syncthreads();
  for (int o = 128; o > 0; o >>= 1) { if (threadIdx.x < o) red[threadIdx.x] += red[threadIdx.x + o]; __syncthreads(); }
  if (threadIdx.x == 0) atomicAdd(&stats[c], red[0]);
  __syncthreads();
  red[threadIdx.x] = s2; __syncthreads();
  for (int o = 128; o > 0; o >>= 1) { if (threadIdx.x < o) red[threadIdx.x] += red[threadIdx.x + o]; __syncthreads(); }
  if (threadIdx.x == 0) atomicAdd(&stats[Cc + c], red[0]);
}

__global__ void bn_final_kernel(const float* __restrict__ stats,
                                const float* __restrict__ g, const float* __restrict__ b,
                                float* __restrict__ scale, float* __restrict__ shift)
{
  int c = blockIdx.x * blockDim.x + threadIdx.x;
  if (c >= Cc) return;
  float n = (float)((long long)Bn * Ss);
  float mu = stats[c] / n;
  float var = stats[Cc + c] / n - mu * mu;
  float sc = g[c] * rsqrtf(var + 1e-5f);
  scale[c] = sc;
  shift[c] = b[c] - mu * sc;
}

// img [B,C,S] -> Xtok [B*S, C]
__global__ void transpose_tokens(const float* __restrict__ img, float* __restrict__ X)
{
  __shared__ float tile[32][33];
  int b = blockIdx.z;
  int s0 = blockIdx.x << 5, c0 = blockIdx.y << 5;
  int tx = threadIdx.x, ty = threadIdx.y;
  for (int i = ty; i < 32; i += 8)
    tile[i][tx] = img[((long long)(b * Cc + c0 + i)) * Ss + s0 + tx];
  __syncthreads();
  for (int i = ty; i < 32; i += 8)
    X[((long long)(b * Ss + s0 + i)) * Cc + c0 + tx] = tile[tx][i];
}

// idx[m] = argmax_p dot(X[m,:], R[p,:])   (first-max tie rule, matches jnp.argmax)
__global__ void route_argmax(const float* __restrict__ X, const float* __restrict__ R,
                             int* __restrict__ idx, int M, int D, int P)
{
  int gv = (blockIdx.x * blockDim.x + threadIdx.x) >> 5;
  int lane = threadIdx.x & 31;
  if (gv >= M) return;
  const float* x = X + (long long)gv * D;
  float s[Pp];
  for (int p = 0; p < P; ++p) s[p] = 0.f;
  for (int d = lane; d < D; d += 32) {
    float xv = x[d];
    for (int p = 0; p < P; ++p) s[p] += xv * R[p * D + d];
  }
  int best = 0; float bv = -3.4e38f;
  for (int p = 0; p < P; ++p) {
    float t = wave_sum(s[p]);
    if (t > bv) { bv = t; best = p; }
  }
  if (lane == 0) idx[gv] = best;
}

// measure_kernel: in-place per (token,head) vector of length HD inside [M, C] buffer
__global__ void measure_kernel(float* __restrict__ buf, int M,
                               const float* __restrict__ R, const float* __restrict__ G)
{
  int gv = (blockIdx.x * blockDim.x + threadIdx.x) >> 5;
  int lane = threadIdx.x & 31;
  if (gv >= M * NH) return;
  int m = gv >> 1, h = gv & 1;
  float* z = buf + (long long)m * Cc + h * HD;
  float zv[5];
#pragma unroll
  for (int j = 0; j < 5; ++j) zv[j] = z[lane + 32 * j];
  float sc[Ff];
#pragma unroll
  for (int f = 0; f < Ff; ++f) {
    float a = 0.f;
#pragma unroll
    for (int j = 0; j < 5; ++j) a += zv[j] * R[f * HD + lane + 32 * j];
    sc[f] = wave_sum(a);
  }
  int best = 0;
#pragma unroll
  for (int f = 1; f < Ff; ++f) if (sc[f] > sc[best]) best = f;
  float gamma = fmaxf(G[best], 1.0f);
  float n1 = 0.f, n2 = 0.f, zw[5];
#pragma unroll
  for (int j = 0; j < 5; ++j) {
    float x = zv[j];
    float sp = fmaxf(x, 0.f) + log1pf(expf(-fabsf(x)));   // stable softplus
    n1 += sp * sp;
    float pw = powf(sp, gamma);
    zw[j] = pw;
    n2 += pw * pw;
  }
  n1 = wave_sum(n1); n2 = wave_sum(n2);
  float ratio = fmaxf(sqrtf(n1), 1e-8f) / fmaxf(sqrtf(n2), 1e-8f);
#pragma unroll
  for (int j = 0; j < 5; ++j) z[lane + 32 * j] = zw[j] * ratio;
}

// token differential operator: Qt <- Qt - lambda[argmax_f dot(cat(Qt,Qpt), R[f])] * Qpt
__global__ void tdo_diff_kernel(float* __restrict__ Qt, const float* __restrict__ Qpt, int M,
                                const float* __restrict__ R, const float* __restrict__ lam)
{
  int gv = (blockIdx.x * blockDim.x + threadIdx.x) >> 5;
  int lane = threadIdx.x & 31;
  if (gv >= M * NH) return;
  int m = gv >> 1, h = gv & 1;
  long long base = (long long)m * Cc + h * HD;
  float a[5], b[5];
#pragma unroll
  for (int j = 0; j < 5; ++j) { a[j] = Qt[base + lane + 32 * j]; b[j] = Qpt[base + lane + 32 * j]; }
  float sc[Ff];
#pragma unroll
  for (int f = 0; f < Ff; ++f) {
    float acc = 0.f;
#pragma unroll
    for (int j = 0; j < 5; ++j) {
      int d = lane + 32 * j;
      acc += a[j] * R[f * (2 * HD) + d] + b[j] * R[f * (2 * HD) + HD + d];
    }
    sc[f] = wave_sum(acc);
  }
  int best = 0;
#pragma unroll
  for (int f = 1; f < Ff; ++f) if (sc[f] > sc[best]) best = f;
  float l = lam[best];
#pragma unroll
  for (int j = 0; j < 5; ++j) Qt[base + lane + 32 * j] = a[j] - l * b[j];
}

// out[b,c,s] = Etmp[(b*S+s)*C + c] + dwc[b,c,s]*scale[c] + shift[c]
__global__ void final_out_kernel(const float* __restrict__ E, const float* __restrict__ dwc,
                                 const float* __restrict__ scale, const float* __restrict__ shift,
                                 float* __restrict__ out)
{
  __shared__ float tile[32][33];
  int b = blockIdx.z;
  int s0 = blockIdx.x << 5, c0 = blockIdx.y << 5;
  int tx = threadIdx.x, ty = threadIdx.y;
  for (int i = ty; i < 32; i += 8)
    tile[i][tx] = E[((long long)(b * Ss + s0 + i)) * Cc + c0 + tx];
  __syncthreads();
  for (int i = ty; i < 32; i += 8) {
    int c = c0 + i;
    long long o = ((long long)(b * Cc + c)) * Ss + s0 + tx;
    out[o] = tile[tx][i] + dwc[o] * scale[c] + shift[c];
  }
}

extern "C" void kernel_launch(void* const* d_in, const int* in_sizes, int n_in,
                              void* d_out, int out_size, void* d_ws, size_t ws_size,
                              hipStream_t stream)
{
  (void)in_sizes; (void)n_in; (void)out_size; (void)ws_size;
  const float* img   = (const float*)d_in[0];
  const float* txt   = (const float*)d_in[1];
  const float* Wq    = (const float*)d_in[2];
  const float* Wk    = (const float*)d_in[3];
  const float* Wv    = (const float*)d_in[4];
  const float* Wqb   = (const float*)d_in[5];
  const float* Wkb   = (const float*)d_in[6];
  const float* r_q   = (const float*)d_in[7];
  const float* r_k   = (const float*)d_in[8];
  const float* kqR   = (const float*)d_in[9];
  const float* kqG   = (const float*)d_in[10];
  const float* kkR   = (const float*)d_in[11];
  const float* kkG   = (const float*)d_in[12];
  const float* kqpR  = (const float*)d_in[13];
  const float* kqpG  = (const float*)d_in[14];
  const float* kkpR  = (const float*)d_in[15];
  const float* kkpG  = (const float*)d_in[16];
  const float* tdoRq = (const float*)d_in[17];
  const float* tdoRk = (const float*)d_in[18];
  const float* lmb   = (const float*)d_in[19];
  const float* dwcW  = (const float*)d_in[20];
  const float* bnG   = (const float*)d_in[21];
  const float* bnB   = (const float*)d_in[22];
  const float* Wout  = (const float*)d_in[23];
  float* out = (float*)d_out;

  // workspace layout (floats), buffers reused across phases
  float* ws    = (float*)d_ws;
  float* dwc   = ws;                                   // [B*C*S]
  float* Xt    = dwc + (size_t)BSz * Cc;               // Xtok, later reused as Wout result
  float* Qb    = Xt + (size_t)BSz * Cc;                // Q -> Q_t -> Q_diff
  float* Db    = Qb + (size_t)BSz * Cc;                // Qp -> Qp_t, later reused as O
  float* Kb    = Db + (size_t)BSz * Cc;                // K -> K_t -> K_diff
  float* Vb    = Kb + (size_t)BNz * Cc;                // V
  float* Kp    = Vb + (size_t)BNz * Cc;                // Kp -> Kp_t
  float* KV    = Kp + (size_t)BNz * Cc;                // [B*NH, HD, HD]
  float* stats = KV + (size_t)Bn * NH * HD * HD;       // [2*C]
  float* scl   = stats + 2 * Cc;                       // [C]
  float* shf   = scl + Cc;                             // [C]
  int*   idxq  = (int*)(shf + Cc);                     // [BS]
  int*   idxk  = idxq + BSz;                           // [BN]

  dim3 blk(256);

  // Phase A: depthwise conv + BN stats
  hipMemsetAsync(stats, 0, 2 * Cc * sizeof(float), stream);
  dwc_stats_kernel<<<Bn * Cc, blk, 0, stream>>>(img, dwcW, dwc, stats);
  bn_final_kernel<<<(Cc + 255) / 256, blk, 0, stream>>>(stats, bnG, bnB, scl, shf);

  // Phase B: tokens + routing indices
  transpose_tokens<<<dim3(Ss / 32, Cc / 32, Bn), dim3(32, 8), 0, stream>>>(img, Xt);
  route_argmax<<<(BSz * 32 + 255) / 256, blk, 0, stream>>>(Xt, r_q, idxq, BSz, Cc, Pp);
  route_argmax<<<(BNz * 32 + 255) / 256, blk, 0, stream>>>(txt, r_k, idxk, BNz, Cc, Pp);

  // Phase C: projections (Y = X * W^T)
  wmma_gemm<0, 1><<<dim3(Cc / 64, BSz / 128, 1), blk, 0, stream>>>(
      Xt, Wq, Qb, BSz, Cc, Cc, Cc, Cc, Cc, 0, 0, 0, 0, 0, 0, 1, 1.f, nullptr, 0);
  wmma_gemm<0, 1><<<dim3(Cc / 64, BNz / 128, 1), blk, 0, stream>>>(
      txt, Wk, Kb, BNz, Cc, Cc, Cc, Cc, Cc, 0, 0, 0, 0, 0, 0, 1, 1.f, nullptr, 0);
  wmma_gemm<0, 1><<<dim3(Cc / 64, BNz / 128, 1), blk, 0, stream>>>(
      txt, Wv, Vb, BNz, Cc, Cc, Cc, Cc, Cc, 0, 0, 0, 0, 0, 0, 1, 1.f, nullptr, 0);
  // routed bank projections: masked stores, each row written by exactly one bank
  for (int p = 0; p < Pp; ++p) {
    wmma_gemm<0, 1><<<dim3(Cc / 64, BSz / 128, 1), blk, 0, stream>>>(
        Xt, Wqb + (size_t)p * Cc * Cc, Db, BSz, Cc, Cc, Cc, Cc, Cc,
        0, 0, 0, 0, 0, 0, 1, 1.f, idxq, p);
    wmma_gemm<0, 1><<<dim3(Cc / 64, BNz / 128, 1), blk, 0, stream>>>(
        txt, Wkb + (size_t)p * Cc * Cc, Kp, BNz, Cc, Cc, Cc, Cc, Cc,
        0, 0, 0, 0, 0, 0, 1, 1.f, idxk, p);
  }

  // Phase D: measure kernels (in place)
  measure_kernel<<<(BSz * NH * 32 + 255) / 256, blk, 0, stream>>>(Qb, BSz, kqR, kqG);
  measure_kernel<<<(BNz * NH * 32 + 255) / 256, blk, 0, stream>>>(Kb, BNz, kkR, kkG);
  measure_kernel<<<(BSz * NH * 32 + 255) / 256, blk, 0, stream>>>(Db, BSz, kqpR, kqpG);
  measure_kernel<<<(BNz * NH * 32 + 255) / 256, blk, 0, stream>>>(Kp, BNz, kkpR, kkpG);

  // Phase E: token differential operator -> Q_diff in Qb, K_diff in Kb
  tdo_diff_kernel<<<(BSz * NH * 32 + 255) / 256, blk, 0, stream>>>(Qb, Db, BSz, tdoRq, lmb);
  tdo_diff_kernel<<<(BNz * NH * 32 + 255) / 256, blk, 0, stream>>>(Kb, Kp, BNz, tdoRk, lmb);

  // Phase F: linear attention
  // KV[b,h] = K_diff[b,h]^T (N x hd) * V[b,h] (N x hd)   -> TA=1, TB=0
  wmma_gemm<1, 0><<<dim3((HD + 63) / 64, (HD + 127) / 128, Bn * NH), blk, 0, stream>>>(
      Kb, Vb, KV, HD, HD, Nn, Cc, Cc, HD,
      (long long)Nn * Cc, HD, (long long)Nn * Cc, HD,
      (long long)NH * HD * HD, (long long)HD * HD, NH, 1.f, nullptr, 0);
  // O[b,h] = Q_diff[b,h] (S x hd) * KV[b,h] (hd x hd) * hd^-0.5  -> into Db
  wmma_gemm<0, 0><<<dim3((HD + 63) / 64, Ss / 128, Bn * NH), blk, 0, stream>>>(
      Qb, KV, Db, Ss, HD, HD, Cc, HD, Cc,
      (long long)Ss * Cc, HD, (long long)NH * HD * HD, (long long)HD * HD,
      (long long)Ss * Cc, HD, NH, 1.0f / sqrtf((float)HD), nullptr, 0);

  // Phase G: output projection (reuse Xt), then transpose + BN residual
  wmma_gemm<0, 1><<<dim3(Cc / 64, BSz / 128, 1), blk, 0, stream>>>(
      Db, Wout, Xt, BSz, Cc, Cc, Cc, Cc, Cc, 0, 0, 0, 0, 0, 0, 1, 1.f, nullptr, 0);
  final_out_kernel<<<dim3(Ss / 32, Cc / 32, Bn), dim3(32, 8), 0, stream>>>(Xt, dwc, scl, shf, out);
}